// LIANetLight_57904749085059
// MI455X (gfx1250) — compile-verified
//
#include <hip/hip_runtime.h>
#include <math.h>

// Instant-NGP style multires hash-grid encoding for MI455X (gfx1250).
//
// Memory-bound problem: 64 MiB streamed output (NT stores, keep L2 for the
// table), 33.5M random 8B gathers from a 4 MiB table (fits L2 192MB; warmed
// with global_prefetch). No matmul -> WMMA not applicable; the CDNA5 wins here
// are the cache-control paths (prefetch, temporal hints) and wave32 SALU
// scalarization of the row-uniform hash term.

#define HASH_P1 2654435761u
#define HASH_P2 805459861u

// Warm the feature table into L2 (HT-ish residency) before the gather kernel.
// Emits global_prefetch_b8 on gfx1250; no VGPR return, no counter cost.
__global__ __launch_bounds__(256) void table_prefetch_kernel(const char* __restrict__ p,
                                                             size_t nbytes) {
    size_t off    = ((size_t)blockIdx.x * blockDim.x + threadIdx.x) * 128u;
    size_t stride = (size_t)gridDim.x * blockDim.x * 128u;
    for (; off < nbytes; off += stride) {
        __builtin_prefetch(p + off, 0 /*read*/, 2 /*keep resident*/);
    }
}

// NL > 0: compile-time level count (fully specialized fast path, L==16).
// NL == 0: runtime level count fallback.
// POW2: table size is a power of two -> hash mod becomes an AND mask.
template <int NL, bool POW2>
__global__ __launch_bounds__(256) void hashgrid_encode_kernel(
    const float* __restrict__ x0,        // [B]
    const float* __restrict__ y0,        // [B]
    const float* __restrict__ tables,    // [T, 2]
    const unsigned int* __restrict__ seeds,   // [L]
    const float* __restrict__ level_N,   // [L]
    const int* __restrict__ tile_sz,     // [1] complete_tile_size
    float* __restrict__ out,             // [B, L*2, H, W]
    int W, int n_pix, unsigned int T, unsigned int tmask, int Lrt)
{
    const int b   = blockIdx.y;
    const int pix = blockIdx.x * 256 + (int)threadIdx.x;
    if (pix >= n_pix) return;

    const int yrow = pix / W;          // wave-uniform for W==256, blockDim==256
    const int xcol = pix - yrow * W;

    const float inv_tile = 1.0f / (float)tile_sz[0];   // s_load, wave-uniform
    const float px = (float)xcol + x0[b];
    const float py = (float)yrow + y0[b];              // wave-uniform -> SALU hash term

    const int L = (NL > 0) ? NL : Lrt;
    float* outb = out + (size_t)b * (size_t)(2 * L) * (size_t)n_pix + (size_t)pix;

#pragma unroll 4
    for (int l = 0; l < L; ++l) {
        const float        s    = level_N[l] * inv_tile;  // s_load + s_mul
        const unsigned int seed = seeds[l];               // s_load

        const float xf = px * s;
        const float yf = py * s;
        const float fxf = floorf(xf);
        const float fyf = floorf(yf);
        const float fx = xf - fxf;
        const float fy = yf - fyf;

        // Coordinates are non-negative by construction (x0,y0 >= 0, s > 0).
        const unsigned int ix = (unsigned int)(int)fxf;
        const unsigned int iy = (unsigned int)(int)fyf;

        // Incremental corner hashing: 2 mults total.
        const unsigned int hx0 = ix * HASH_P1;
        const unsigned int hx1 = hx0 + HASH_P1;
        const unsigned int hyb = iy * HASH_P2;
        const unsigned int hy0 = hyb ^ seed;               // uniform -> SGPR
        const unsigned int hy1 = (hyb + HASH_P2) ^ seed;   // uniform -> SGPR

        unsigned int h00 = hx0 ^ hy0;
        unsigned int h10 = hx1 ^ hy0;
        unsigned int h01 = hx0 ^ hy1;
        unsigned int h11 = hx1 ^ hy1;
        if (POW2) {
            h00 &= tmask; h10 &= tmask; h01 &= tmask; h11 &= tmask;
        } else {
            h00 %= T;     h10 %= T;     h01 %= T;     h11 %= T;
        }

        // 2 adjacent f32 features per corner -> single global_load_b64 each.
        const float2 f00 = *reinterpret_cast<const float2*>(tables + 2u * h00);
        const float2 f10 = *reinterpret_cast<const float2*>(tables + 2u * h10);
        const float2 f01 = *reinterpret_cast<const float2*>(tables + 2u * h01);
        const float2 f11 = *reinterpret_cast<const float2*>(tables + 2u * h11);

        const float gx = 1.0f - fx;
        const float gy = 1.0f - fy;
        const float w00 = gx * gy;
        const float w10 = fx * gy;
        const float w01 = gx * fy;
        const float w11 = fx * fy;

        const float e0 = fmaf(w11, f11.x, fmaf(w01, f01.x, fmaf(w10, f10.x, w00 * f00.x)));
        const float e1 = fmaf(w11, f11.y, fmaf(w01, f01.y, fmaf(w10, f10.y, w00 * f00.y)));

        // Write-once 64 MiB stream: non-temporal so the hash table stays in L2.
        __builtin_nontemporal_store(e0, outb + (size_t)(2 * l)     * (size_t)n_pix);
        __builtin_nontemporal_store(e1, outb + (size_t)(2 * l + 1) * (size_t)n_pix);
    }
}

extern "C" void kernel_launch(void* const* d_in, const int* in_sizes, int n_in,
                              void* d_out, int out_size, void* d_ws, size_t ws_size,
                              hipStream_t stream) {
    (void)n_in; (void)d_ws; (void)ws_size;

    const float*        x0      = (const float*)d_in[0];
    const float*        y0      = (const float*)d_in[1];
    const float*        tables  = (const float*)d_in[2];
    const unsigned int* seeds   = (const unsigned int*)d_in[3];
    const float*        level_N = (const float*)d_in[4];
    const int*          tile_sz = (const int*)d_in[6];  // complete_tile_size

    const int B = in_sizes[0];
    const int L = in_sizes[3];
    const int F = 2;  // FEAT_DIM in the reference
    const unsigned int T = (unsigned int)(in_sizes[2] / F);
    const int n_pix = out_size / (B * L * F);  // H*W plane size

    // H == W (square crop): integer sqrt of n_pix.
    int W = 1;
    while ((long long)W * W < (long long)n_pix) ++W;

    const bool pow2 = (T != 0u) && ((T & (T - 1u)) == 0u);
    const unsigned int tmask = T - 1u;

    // Warm the 4 MiB table into L2 once per call (cheap: ~0.2us even on miss).
    table_prefetch_kernel<<<64, 256, 0, stream>>>(
        (const char*)tables, (size_t)in_sizes[2] * sizeof(float));

    dim3 grid((unsigned)((n_pix + 255) / 256), (unsigned)B);
    dim3 block(256);
    float* out = (float*)d_out;

    if (L == 16) {
        if (pow2) {
            hashgrid_encode_kernel<16, true><<<grid, block, 0, stream>>>(
                x0, y0, tables, seeds, level_N, tile_sz, out, W, n_pix, T, tmask, L);
        } else {
            hashgrid_encode_kernel<16, false><<<grid, block, 0, stream>>>(
                x0, y0, tables, seeds, level_N, tile_sz, out, W, n_pix, T, tmask, L);
        }
    } else {
        if (pow2) {
            hashgrid_encode_kernel<0, true><<<grid, block, 0, stream>>>(
                x0, y0, tables, seeds, level_N, tile_sz, out, W, n_pix, T, tmask, L);
        } else {
            hashgrid_encode_kernel<0, false><<<grid, block, 0, stream>>>(
                x0, y0, tables, seeds, level_N, tile_sz, out, W, n_pix, T, tmask, L);
        }
    }
}